// LSTMGCN_10264971838232
// MI455X (gfx1250) — compile-verified
//
#include <hip/hip_runtime.h>
#include <math.h>

typedef float v2f __attribute__((ext_vector_type(2)));
typedef float v8f __attribute__((ext_vector_type(8)));

#define F            32     // node feature width (in == out)
#define KDIM         64     // concat(x, tx1) contraction dim
#define GCOLS        128    // 4 gates * 32
#define TILE_NODES   64     // nodes per GEMM block
#define GEMM_THREADS 128    // 4 wave32
#define SWP_STRIDE   144    // float2 pairs per K-pair row (bank-shift 32 for hi half)
#define SIN_STRIDE   66     // even -> 8B-aligned A-pair loads

__device__ __forceinline__ float sigmoidf_(float x) {
    return 1.0f / (1.0f + expf(-x));
}

__device__ __forceinline__ void atomic_fadd(float* p, float v) {
    __hip_atomic_fetch_add(p, v, __ATOMIC_RELAXED, __HIP_MEMORY_SCOPE_AGENT);
}

// ---- init: zero deg & tx1, build concatenated weight [64x128] and fused bias ----
__global__ void k_init(float* __restrict__ deg, float* __restrict__ tx1,
                       float* __restrict__ Wcat, float* __restrict__ biasCat,
                       const float* __restrict__ Wx, const float* __restrict__ bx,
                       const float* __restrict__ bh, const float* __restrict__ bg,
                       int n) {
    int tid = blockIdx.x * blockDim.x + threadIdx.x;
    int stride = gridDim.x * blockDim.x;
    for (int i = tid; i < n; i += stride) deg[i] = 0.0f;
    long tot = (long)n * F;
    for (long i = tid; i < tot; i += stride) tx1[i] = 0.0f;
    for (int i = tid; i < KDIM * GCOLS; i += stride) {
        int k = i >> 7;       // 0..63
        int c = i & 127;      // gate*32 + j
        int g = c >> 5;
        int j = c & 31;
        // Wx shape (4, K=2, 32, 32) row-major
        float v = (k < F) ? Wx[((g * 2 + 0) * F + k) * F + j]
                          : Wx[((g * 2 + 1) * F + (k - F)) * F + j];
        Wcat[i] = v;
    }
    for (int i = tid; i < GCOLS; i += stride)
        biasCat[i] = bx[i] + bh[i] + bg[i];  // cheb(H=0,...) == bh
}

// ---- degree accumulation (self-loops zeroed, sum over src) ----
__global__ void k_deg(const long long* __restrict__ ei, const float* __restrict__ ew,
                      float* __restrict__ deg, int e) {
    int i = blockIdx.x * blockDim.x + threadIdx.x;
    if (i >= e) return;
    int s = (int)ei[i];
    int d = (int)ei[(long)e + i];
    if (s != d) atomic_fadd(&deg[s], ew[i]);
}

__global__ void k_dinv(const float* __restrict__ deg, float* __restrict__ dinv, int n) {
    int i = blockIdx.x * blockDim.x + threadIdx.x;
    if (i >= n) return;
    float dg = deg[i];
    dinv[i] = (dg > 0.0f) ? rsqrtf(dg) : 0.0f;
}

// ---- scatter SpMM: tx1[dst,:] += (-dinv[src]*w*dinv[dst]) * x[src,:]; one wave32/edge ----
__global__ void k_spmm(const long long* __restrict__ ei, const float* __restrict__ ew,
                       const float* __restrict__ dinv, const float* __restrict__ x,
                       float* __restrict__ tx1, int e) {
    int eidx = blockIdx.x * 8 + (threadIdx.x >> 5);
    int lane = threadIdx.x & 31;
    if (eidx >= e) return;
    int s = (int)ei[eidx];
    int d = (int)ei[(long)e + eidx];
    if (s == d) return;                       // self-loop weight zeroed
    float coeff = -dinv[s] * ew[eidx] * dinv[d];
    float xv = x[(long)s * F + lane];
    atomic_fadd(&tx1[(long)d * F + lane], coeff * xv);
}

// ---- WMMA GEMM ([64 nodes x 64] @ [64 x 128]) + fused LSTM elementwise + output dot ----
__global__ __launch_bounds__(GEMM_THREADS)
void k_gemm(const float* __restrict__ x, const float* __restrict__ tx1,
            const float* __restrict__ Wcat, const float* __restrict__ biasCat,
            const float* __restrict__ wc, const float* __restrict__ lin_w,
            const float* __restrict__ lin_b, float* __restrict__ out, int n) {
    // Weights stored as interleaved K-pairs: pair q = k/2, element {W[2q][col], W[2q+1][col]}.
    // One aligned ds_load_b64 yields a ready-to-use even-aligned B fragment pair.
    __shared__ float sWp[2 * (KDIM / 2) * SWP_STRIDE];   // 36864 B
    __shared__ float sIn[TILE_NODES * SIN_STRIDE];       // 16896 B

    int t = threadIdx.x;
    int rowBase = blockIdx.x * TILE_NODES;

    // stage weights: interleave consecutive-K pairs per column
    for (int i = t; i < (KDIM / 2) * GCOLS; i += GEMM_THREADS) {
        int p = i >> 7;          // K-pair index 0..31
        int col = i & 127;
        v2f w;
        w.x = Wcat[(2 * p + 0) * GCOLS + col];
        w.y = Wcat[(2 * p + 1) * GCOLS + col];
        *(v2f*)&sWp[2 * (p * SWP_STRIDE + col)] = w;
    }

    // stage concat(x, tx1) rows for this block's nodes (coalesced reads)
    for (int i = t; i < TILE_NODES * F; i += GEMM_THREADS) {
        int r = i >> 5, k = i & 31;
        int g = rowBase + r;
        float xv = 0.0f, tv = 0.0f;
        if (g < n) { xv = x[(long)g * F + k]; tv = tx1[(long)g * F + k]; }
        sIn[r * SIN_STRIDE + k]     = xv;
        sIn[r * SIN_STRIDE + F + k] = tv;
    }
    __syncthreads();

    int wave   = t >> 5;         // 0..3
    int lane   = t & 31;
    int laneLo = lane & 15;
    int hi     = lane >> 4;      // 0 or 1
    int wRow   = wave * 16;      // local node-row base for this wave

    // A fragments: lanes<16 hold K={kb,kb+1}, lanes>=16 hold {kb+2,kb+3}; single b64 loads.
    const float* abase = &sIn[(wRow + laneLo) * SIN_STRIDE + hi * 2];
    v2f a[16];
#pragma unroll
    for (int kt = 0; kt < 16; ++kt)
        a[kt] = *(const v2f*)(abase + 4 * kt);

    // 6 output column tiles: gate i (cols 0..31), gate c (64..95), gate o (96..127).
    // Forget gate (cols 32..63) is dead because C_prev == 0.
    v8f acc[6];
#pragma unroll
    for (int tt = 0; tt < 6; ++tt) {
        int ct  = (tt < 2) ? tt : tt + 2;       // actual column tile 0,1,4,5,6,7
        int col = ct * 16 + laneLo;
        float b0 = biasCat[col];
        v8f c;
#pragma unroll
        for (int i = 0; i < 8; ++i) c[i] = b0;  // C starts as fused bias
        const float* wbase = &sWp[2 * (hi * SWP_STRIDE + col)];
#pragma unroll
        for (int kt = 0; kt < 16; ++kt) {
            // pair q = kt*2 + hi  ->  float offset 2*q*SWP_STRIDE from (hi,col) base
            v2f b = *(const v2f*)(wbase + 4 * SWP_STRIDE * kt);
            c = __builtin_amdgcn_wmma_f32_16x16x4_f32(
                    false, a[kt], false, b, (short)0, c, false, false);
        }
        acc[tt] = c;
    }

    // Elementwise LSTM in fragment layout: element (vgpr i, lane) = node rowBase+wRow+i+8*hi,
    // feature f = h*16 + laneLo (h = tile half within gate).
    float partial[8];
#pragma unroll
    for (int i = 0; i < 8; ++i) partial[i] = 0.0f;

#pragma unroll
    for (int h = 0; h < 2; ++h) {
        int f = h * 16 + laneLo;
        float wc2f = wc[2 * F + f];       // peephole on output gate
        float lwf  = lin_w[f];
#pragma unroll
        for (int i = 0; i < 8; ++i) {
            float zi = acc[h][i];
            float zc = acc[2 + h][i];
            float zo = acc[4 + h][i];
            float cv = sigmoidf_(zi) * tanhf(zc);          // C = I * T
            float ov = sigmoidf_(zo + wc2f * cv);          // O
            float hv = ov * tanhf(cv);                     // H
            partial[i] += fmaxf(hv, 0.0f) * lwf;           // relu(H) . lin_w
        }
    }

    // Reduce the 32-feature dot across each 16-lane half (masks 1,2,4,8 stay inside halves)
    float lb = lin_b[0];
#pragma unroll
    for (int i = 0; i < 8; ++i) {
        float v = partial[i];
#pragma unroll
        for (int m = 1; m < 16; m <<= 1) v += __shfl_xor(v, m, 32);
        if (laneLo == 0) {
            int node = rowBase + wRow + i + 8 * hi;
            if (node < n) out[node] = v + lb;
        }
    }
}

extern "C" void kernel_launch(void* const* d_in, const int* in_sizes, int n_in,
                              void* d_out, int out_size, void* d_ws, size_t ws_size,
                              hipStream_t stream) {
    const float*     x     = (const float*)d_in[0];
    const long long* ei    = (const long long*)d_in[1];   // int64 (2, E)
    const float*     ew    = (const float*)d_in[2];
    const float*     Wx    = (const float*)d_in[3];       // (4,2,32,32)
    const float*     bx    = (const float*)d_in[4];       // (4,32)
    /* Wh = d_in[5] unused: H_prev == 0 */
    const float*     bh    = (const float*)d_in[6];       // (4,32)
    const float*     wc    = (const float*)d_in[7];       // (3,32)
    const float*     bg    = (const float*)d_in[8];       // (4,32)
    const float*     lin_w = (const float*)d_in[9];       // (32,1)
    const float*     lin_b = (const float*)d_in[10];      // (1,)
    float* out = (float*)d_out;

    int n = in_sizes[0] / F;     // 100000
    int e = in_sizes[2];         // 1600000

    float* ws      = (float*)d_ws;
    float* deg     = ws;                    // n
    float* dinv    = deg + n;               // n
    float* Wcat    = dinv + n;              // 64*128
    float* biasCat = Wcat + KDIM * GCOLS;   // 128
    float* tx1     = biasCat + GCOLS;       // n*32

    k_init<<<1024, 256, 0, stream>>>(deg, tx1, Wcat, biasCat, Wx, bx, bh, bg, n);
    k_deg<<<(e + 255) / 256, 256, 0, stream>>>(ei, ew, deg, e);
    k_dinv<<<(n + 255) / 256, 256, 0, stream>>>(deg, dinv, n);
    k_spmm<<<(e + 7) / 8, 256, 0, stream>>>(ei, ew, dinv, x, tx1, e);
    k_gemm<<<(n + TILE_NODES - 1) / TILE_NODES, GEMM_THREADS, 0, stream>>>(
        x, tx1, Wcat, biasCat, wc, lin_w, lin_b, out, n);
}